// PDFSampler_11029476016212
// MI455X (gfx1250) — compile-verified
//
#include <hip/hip_runtime.h>
#include <cstdint>

#define EPS 1e-5f
#define RAYS_PER_WG 128
#define WG_THREADS 256   // 8 waves; each wave owns 16 rays for the WMMA phase

typedef __attribute__((ext_vector_type(2))) float v2f;
typedef __attribute__((ext_vector_type(4))) float v4f;
typedef __attribute__((ext_vector_type(8))) float v8f;
typedef __attribute__((ext_vector_type(4))) int   v4i;

// ---- probe async global->LDS path (CDNA5 ASYNCcnt ops), fall back cleanly ----
#if defined(__has_builtin)
#if __has_builtin(__builtin_amdgcn_global_load_async_to_lds_b128)
#define HAVE_ASYNC_B128 1
#endif
#if __has_builtin(__builtin_amdgcn_s_wait_asynccnt)
#define HAVE_WAIT_ASYNC 1
#endif
#endif

typedef __attribute__((address_space(1))) v4i gas_v4i;   // global int4
typedef __attribute__((address_space(3))) v4i las_v4i;   // LDS int4

__global__ __launch_bounds__(WG_THREADS) void pdf_sampler_kernel(
    const float* __restrict__ gbins,     // [R,65]
    const float* __restrict__ gweights,  // [R,64]
    const float* __restrict__ gu,        // [R,64]
    float* __restrict__ gout)            // [R,64]
{
  __shared__ float lds_cdf [RAYS_PER_WG * 65 + 4];
  __shared__ float lds_bins[RAYS_PER_WG * 65 + 4];

  const int  tid  = (int)threadIdx.x;
  const int  lane = tid & 31;
  const int  wave = tid >> 5;
  const int  hl   = lane >> 4;      // 0: lanes 0-15, 1: lanes 16-31
  const int  nco  = lane & 15;      // row-in-tile (A) / column (B,C,D)
  const long ray0 = (long)blockIdx.x * RAYS_PER_WG;

  // ---------------- stage bins[128][65] into LDS (coalesced b128) ----------------
  {
    const float* src = gbins + ray0 * 65;   // block base: 33280 B -> 16B aligned
#if HAVE_ASYNC_B128
    for (int i = tid; i < (RAYS_PER_WG * 65) / 4; i += WG_THREADS) {
      __builtin_amdgcn_global_load_async_to_lds_b128(
          (gas_v4i*)(src + 4 * i), (las_v4i*)(lds_bins + 4 * i), 0, 0);
    }
#else
    for (int i = tid; i < (RAYS_PER_WG * 65) / 4; i += WG_THREADS) {
      *(v4f*)&lds_bins[4 * i] = *(const v4f*)&src[4 * i];
    }
#endif
  }

  // prefetch the u cacheline this thread will consume later (global_prefetch_b8)
  __builtin_prefetch(gu + ray0 * 64 + (long)tid * 32, 0, 0);

  // ---------------- WMMA prefix-sum of raw weights (16 rays per wave) ------------
  // S[r, j] = sum_k w[r,k] * T[k,j],  T[k,j] = (k <= j)
  const int  rtile = wave * 16;
  const float* wrow = gweights + (ray0 + rtile + nco) * 64 + hl * 2;

  // A fragments, 16 K-chunks of 4 (f32 16x4 layout: VGPR0={K0|K2}, VGPR1={K1|K3})
  v2f a[16];
#pragma unroll
  for (int c = 0; c < 16; ++c)
    a[c] = *(const v2f*)(wrow + 4 * c);

  v8f acc[4] = {};
#pragma unroll
  for (int jb = 0; jb < 4; ++jb) {
    const int jg = 16 * jb + nco;            // global column for this lane
#pragma unroll
    for (int c = 0; c < 4 * (jb + 1); ++c) { // only K-blocks with k <= jmax
      const int k0 = 4 * c + 2 * hl;         // B 4x16 layout mirrors A striping
      v2f b;
      b.x = (k0     <= jg) ? 1.0f : 0.0f;
      b.y = (k0 + 1 <= jg) ? 1.0f : 0.0f;
      acc[jb] = __builtin_amdgcn_wmma_f32_16x16x4_f32(
          false, a[c], false, b, (short)0, acc[jb], false, false);
    }
  }

  // w_sum[r] = S[r,63] lives at N=15 -> lanes 15/31 of acc[3]; broadcast per half
  const int bidx = (lane | 15) << 2;
  float cpad[8], crd[8];
#pragma unroll
  for (int v = 0; v < 8; ++v) {
    float ws = __int_as_float(
        __builtin_amdgcn_ds_bpermute(bidx, __float_as_int(acc[3][v])));
    float pd = fmaxf(EPS - ws, 0.0f);        // relu(eps - w_sum)
    cpad[v] = pd;
    crd[v]  = 1.0f / (ws + pd);
  }

  // normalized cdf -> LDS: cdf[r][1+j] = min(1, (S + pad*(j+1)/64) / (wsum+pad))
#pragma unroll
  for (int jb = 0; jb < 4; ++jb) {
    const int   jg  = 16 * jb + nco;
    const float jp1 = (float)(jg + 1) * (1.0f / 64.0f);
#pragma unroll
    for (int v = 0; v < 8; ++v) {
      const int rloc = rtile + v + hl * 8;   // C/D layout: M = v (+8 upper half)
      float cv = fminf(1.0f, (acc[jb][v] + cpad[v] * jp1) * crd[v]);
      lds_cdf[rloc * 65 + 1 + jg] = cv;
    }
  }
  if (tid < RAYS_PER_WG) lds_cdf[tid * 65] = 0.0f;   // leading zero

#if HAVE_ASYNC_B128
#if HAVE_WAIT_ASYNC
  __builtin_amdgcn_s_wait_asynccnt(0);
#else
  asm volatile("s_wait_asynccnt 0" ::: "memory");
#endif
#endif
  __syncthreads();

  // ---------------- inverse-CDF search + lerp (2 threads per ray) ----------------
  const int   rloc = tid >> 1;
  const int   j0   = (tid & 1) * 32;
  const long  rg   = ray0 + rloc;
  const float* urow = gu   + rg * 64 + j0;
  float*       orow = gout + rg * 64 + j0;
  const float* cdf  = &lds_cdf [rloc * 65];
  const float* bn   = &lds_bins[rloc * 65];

#pragma unroll
  for (int q = 0; q < 8; ++q) {
    v4f uv = *(const v4f*)(urow + 4 * q);
    v4f res;
#pragma unroll
    for (int k = 0; k < 4; ++k) {
      const float uu = uv[k];
      // largest i in [0,64] with cdf[i] <= uu (cdf[0]=0 <= uu always)
      int i = 0;
#pragma unroll
      for (int st = 64; st >= 1; st >>= 1) {
        int cand = i + st;
        i = (cand <= 64 && cdf[cand] <= uu) ? cand : i;
      }
      const int ab = (i + 1 > 64) ? 64 : i + 1;   // 'above' clamp
      const float c0 = cdf[i];
      const float c1 = cdf[ab];
      float d = c1 - c0;
      d = (d < EPS) ? 1.0f : d;
      const float t  = (uu - c0) / d;
      const float b0 = bn[i];
      res[k] = b0 + t * (bn[ab] - b0);
    }
    *(v4f*)(orow + 4 * q) = res;
  }
}

extern "C" void kernel_launch(void* const* d_in, const int* in_sizes, int n_in,
                              void* d_out, int out_size, void* d_ws, size_t ws_size,
                              hipStream_t stream) {
  const float* bins    = (const float*)d_in[0];
  const float* weights = (const float*)d_in[1];
  const float* u       = (const float*)d_in[2];
  float*       out     = (float*)d_out;

  const int R    = in_sizes[1] / 64;            // weights is [R,64]
  const int grid = R / RAYS_PER_WG;             // R = 524288 -> 4096 blocks
  pdf_sampler_kernel<<<grid, WG_THREADS, 0, stream>>>(bins, weights, u, out);
}